// EnhancedGNN_83571473645725
// MI455X (gfx1250) — compile-verified
//
#include <hip/hip_runtime.h>
#include <hip/hip_bf16.h>

#define N_NODES 50000
#define N_EDGES 400000
#define F_INDIM 10
#define H 128
#define H2 256
#define HEADS 4
#define ED 5
#define GAT_C (HEADS * H) /* 512 */
#define BN_EPS 1e-5f

typedef float v2f __attribute__((ext_vector_type(2)));
typedef float v8f __attribute__((ext_vector_type(8)));

__device__ __forceinline__ v8f wmma_f32x4(v2f a, v2f b, v8f c) {
  // V_WMMA_F32_16X16X4_F32 : D = A(16x4,f32) x B(4x16,f32) + C(16x16,f32)
  return __builtin_amdgcn_wmma_f32_16x16x4_f32(false, a, false, b, (short)0, c,
                                               false, false);
}

__device__ __forceinline__ void atomicMaxFloat(float* addr, float val) {
  // monotonic-bits trick: works for mixed-sign floats, init = -3e38
  if (val >= 0.f)
    atomicMax((int*)addr, __float_as_int(val));
  else
    atomicMin((unsigned int*)addr, (unsigned int)__float_as_int(val));
}

// ---------------------------------------------------------------- fill
__global__ void fill_kernel(float* __restrict__ p, long n, float v) {
  long i = (long)blockIdx.x * blockDim.x + threadIdx.x;
  if (i < n) p[i] = v;
}

// ---------------------------------------------------------------- input proj
// h0 = relu(BN(x @ w_in + b_in)); 2 rows per 256-thread block
__global__ __launch_bounds__(256) void input_proj_kernel(
    const float* __restrict__ x, const float* __restrict__ w_in,
    const float* __restrict__ b_in, const float* __restrict__ g,
    const float* __restrict__ b, const float* __restrict__ m,
    const float* __restrict__ v, float* __restrict__ h0) {
  __shared__ float xs[2][F_INDIM];
  int row0 = blockIdx.x * 2;
  int t = threadIdx.x;
  if (t < 2 * F_INDIM)
    xs[t / F_INDIM][t % F_INDIM] = x[(long)(row0 + t / F_INDIM) * F_INDIM + t % F_INDIM];
  __syncthreads();
  int r = t >> 7;
  int c = t & 127;
  float acc = b_in[c];
#pragma unroll
  for (int k = 0; k < F_INDIM; ++k) acc += xs[r][k] * w_in[k * H + c];
  float bn = (acc - m[c]) * g[c] * rsqrtf(v[c] + BN_EPS) + b[c];
  h0[(long)(row0 + r) * H + c] = fmaxf(bn, 0.f);
}

// ---------------------------------------------------------------- GINE edges
// agg[dst] += relu(h0[src] + edge_attr @ gine_ew + gine_eb)
__global__ __launch_bounds__(256) void gine_edge_kernel(
    const int* __restrict__ src, const int* __restrict__ dst,
    const float* __restrict__ edge_attr, const float* __restrict__ ew,
    const float* __restrict__ eb, const float* __restrict__ h0,
    float* __restrict__ agg) {
  int t = threadIdx.x;
  int e = blockIdx.x * 2 + (t >> 7);
  int c = t & 127;
  const float* ea = edge_attr + (long)e * ED;
  float em = eb[c];
#pragma unroll
  for (int k = 0; k < ED; ++k) em += ea[k] * ew[k * H + c];
  int s = src[e], d = dst[e];
  float msg = fmaxf(h0[(long)s * H + c] + em, 0.f);
  atomicAdd(&agg[(long)d * H + c], msg);
}

// ---------------------------------------------------------------- fused GINE MLP
// x1 = BN1(s0*h0 + (1-s0)*relu( relu(BN(z@w1+b1)) @ w2 + b2 )), z = h0+agg
// block = 256 threads (8 waves), 16 rows per block, both GEMMs via f32 WMMA
__global__ __launch_bounds__(256) void gine_mlp_kernel(
    const float* __restrict__ h0, const float* __restrict__ agg,
    const float* __restrict__ w1, const float* __restrict__ b1,
    const float* __restrict__ bng, const float* __restrict__ bnb,
    const float* __restrict__ bnm, const float* __restrict__ bnv,
    const float* __restrict__ w2, const float* __restrict__ b2,
    const float* __restrict__ bn1g, const float* __restrict__ bn1b,
    const float* __restrict__ bn1m, const float* __restrict__ bn1v,
    const float* __restrict__ skip0, float* __restrict__ x1) {
  __shared__ float As[16 * 132];  // z tile, padded stride
  __shared__ float Ts[16 * 260];  // hidden tile, padded stride
  int r0 = blockIdx.x * 16;
  int t = threadIdx.x;
  for (int i = t; i < 16 * H; i += 256) {
    int mm = i >> 7, kk = i & 127;
    long gi = (long)(r0 + mm) * H + kk;
    As[mm * 132 + kk] = h0[gi] + agg[gi];
  }
  __syncthreads();
  int wave = t >> 5, lane = t & 31;
  int lh = lane >> 4, lmod = lane & 15;
  // ---- GEMM1: z(16x128) @ w1(128x256) -> BN+ReLU -> Ts
  for (int tile = wave; tile < H2 / 16; tile += 8) {
    int n = tile * 16 + lmod;
    v8f acc = {};
    for (int k = 0; k < H; k += 4) {
      int ka = k + 2 * lh;
      v2f a;
      a.x = As[lmod * 132 + ka];
      a.y = As[lmod * 132 + ka + 1];
      v2f bb;
      bb.x = w1[(long)ka * H2 + n];
      bb.y = w1[(long)(ka + 1) * H2 + n];
      acc = wmma_f32x4(a, bb, acc);
    }
    float scale = bng[n] * rsqrtf(bnv[n] + BN_EPS);
    float mean = bnm[n], beta = bnb[n], bias = b1[n];
#pragma unroll
    for (int i = 0; i < 8; ++i) {
      float vv = (acc[i] + bias - mean) * scale + beta;
      Ts[(lh * 8 + i) * 260 + n] = fmaxf(vv, 0.f);
    }
  }
  __syncthreads();
  // ---- GEMM2: Ts(16x256) @ w2(256x128) -> relu -> skip-mix -> BN1 -> x1
  float s0 = 1.f / (1.f + __expf(-skip0[0]));
  {
    int n = wave * 16 + lmod;
    v8f acc = {};
    for (int k = 0; k < H2; k += 4) {
      int ka = k + 2 * lh;
      v2f a;
      a.x = Ts[lmod * 260 + ka];
      a.y = Ts[lmod * 260 + ka + 1];
      v2f bb;
      bb.x = w2[(long)ka * H + n];
      bb.y = w2[(long)(ka + 1) * H + n];
      acc = wmma_f32x4(a, bb, acc);
    }
    float scale = bn1g[n] * rsqrtf(bn1v[n] + BN_EPS);
    float mean = bn1m[n], beta = bn1b[n], bias = b2[n];
#pragma unroll
    for (int i = 0; i < 8; ++i) {
      int row = r0 + lh * 8 + i;
      float xnew = fmaxf(acc[i] + bias, 0.f);
      float mix = s0 * h0[(long)row * H + n] + (1.f - s0) * xnew;
      x1[(long)row * H + n] = (mix - mean) * scale + beta;
    }
  }
}

// ---------------------------------------------------------------- GAT node transforms
// xl = x1@wl + bl ; xr = x1@wr + br   (both [N, 512]) via f32 WMMA
__global__ __launch_bounds__(256) void gat_transform_kernel(
    const float* __restrict__ x1, const float* __restrict__ wl,
    const float* __restrict__ bl, const float* __restrict__ wr,
    const float* __restrict__ br, float* __restrict__ xl,
    float* __restrict__ xr) {
  __shared__ float As[16 * 132];
  int r0 = blockIdx.x * 16;
  int t = threadIdx.x;
  for (int i = t; i < 16 * H; i += 256) {
    int mm = i >> 7, kk = i & 127;
    As[mm * 132 + kk] = x1[(long)(r0 + mm) * H + kk];
  }
  __syncthreads();
  int wave = t >> 5, lane = t & 31;
  int lh = lane >> 4, lmod = lane & 15;
  for (int tile = wave; tile < 2 * GAT_C / 16; tile += 8) {
    bool left = tile < GAT_C / 16;  // uniform per wave
    const float* W = left ? wl : wr;
    const float* B = left ? bl : br;
    float* O = left ? xl : xr;
    int n = (left ? tile : tile - GAT_C / 16) * 16 + lmod;
    v8f acc = {};
    for (int k = 0; k < H; k += 4) {
      int ka = k + 2 * lh;
      v2f a;
      a.x = As[lmod * 132 + ka];
      a.y = As[lmod * 132 + ka + 1];
      v2f bb;
      bb.x = W[(long)ka * GAT_C + n];
      bb.y = W[(long)(ka + 1) * GAT_C + n];
      acc = wmma_f32x4(a, bb, acc);
    }
    float bias = B[n];
#pragma unroll
    for (int i = 0; i < 8; ++i) {
      int row = r0 + lh * 8 + i;
      O[(long)row * GAT_C + n] = acc[i] + bias;
    }
  }
}

// ---------------------------------------------------------------- GAT scores
// score[e,h] = att[h] . leaky_relu(xl[src]+xr[dst]+edge_attr@we, 0.2)
// one wave per (edge, head); shfl reduction; atomic max into mmax[dst,h]
__global__ __launch_bounds__(256) void gat_score_kernel(
    const int* __restrict__ src, const int* __restrict__ dst,
    const float* __restrict__ edge_attr, const float* __restrict__ we,
    const float* __restrict__ att, const float* __restrict__ xl,
    const float* __restrict__ xr, float* __restrict__ score,
    float* __restrict__ mmax) {
  int t = threadIdx.x;
  int wave = t >> 5, lane = t & 31;
  int e = blockIdx.x * 2 + (wave >> 2);
  int h = wave & 3;
  int s = src[e], d = dst[e];
  float ea[ED];
#pragma unroll
  for (int k = 0; k < ED; ++k) ea[k] = edge_attr[(long)e * ED + k];
  float sum = 0.f;
#pragma unroll
  for (int j = 0; j < 4; ++j) {
    int col = h * H + lane + j * 32;
    float ee = 0.f;
#pragma unroll
    for (int k = 0; k < ED; ++k) ee += ea[k] * we[k * GAT_C + col];
    float val = xl[(long)s * GAT_C + col] + xr[(long)d * GAT_C + col] + ee;
    val = val > 0.f ? val : 0.2f * val;
    sum += val * att[col];
  }
#pragma unroll
  for (int off = 16; off > 0; off >>= 1) sum += __shfl_xor(sum, off, 32);
  if (lane == 0) {
    score[(long)e * HEADS + h] = sum;
    atomicMaxFloat(&mmax[(long)d * HEADS + h], sum);
  }
}

// ---------------------------------------------------------------- softmax exp+sum
__global__ void gat_expsum_kernel(const int* __restrict__ dst,
                                  const float* __restrict__ mmax,
                                  float* __restrict__ score,
                                  float* __restrict__ ssum) {
  long i = (long)blockIdx.x * blockDim.x + threadIdx.x;
  if (i >= (long)N_EDGES * HEADS) return;
  int e = (int)(i >> 2), h = (int)(i & 3);
  int d = dst[e];
  float mm = mmax[(long)d * HEADS + h];
  if (mm < -1e30f) mm = 0.f;
  float ex = __expf(score[i] - mm);
  score[i] = ex;  // store exp for the aggregation pass
  atomicAdd(&ssum[(long)d * HEADS + h], ex);
}

// ---------------------------------------------------------------- weighted aggregation
__global__ __launch_bounds__(256) void gat_aggregate_kernel(
    const int* __restrict__ src, const int* __restrict__ dst,
    const float* __restrict__ score, const float* __restrict__ ssum,
    const float* __restrict__ xl, float* __restrict__ gatout) {
  int t = threadIdx.x;
  int e = blockIdx.x * 2 + (t >> 7);
  int c = t & 127;
  int s = src[e], d = dst[e];
#pragma unroll
  for (int h = 0; h < HEADS; ++h) {
    float alpha =
        score[(long)e * HEADS + h] / (ssum[(long)d * HEADS + h] + 1e-16f);
    float v = alpha * xl[(long)s * GAT_C + h * H + c];
    atomicAdd(&gatout[((long)d * HEADS + h) * H + c], v);
  }
}

// ---------------------------------------------------------------- finalize
__global__ void finalize_kernel(
    const float* __restrict__ h0, const float* __restrict__ x1,
    const float* __restrict__ gatout, const float* __restrict__ gat_bias,
    const float* __restrict__ bn2g, const float* __restrict__ bn2b,
    const float* __restrict__ bn2m, const float* __restrict__ bn2v,
    const float* __restrict__ skip1, float* __restrict__ out) {
  long i = (long)blockIdx.x * blockDim.x + threadIdx.x;
  if (i >= (long)N_NODES * H) return;
  long n = i >> 7;
  int c = (int)(i & 127);
  float gm = 0.f;
#pragma unroll
  for (int h = 0; h < HEADS; ++h) gm += gatout[(n * HEADS + h) * H + c];
  gm = gm * 0.25f + gat_bias[c];
  float xnew = fmaxf(gm, 0.f);
  float s1 = 1.f / (1.f + __expf(-skip1[0]));
  float x1v = x1[i];
  float mix = s1 * x1v + (1.f - s1) * xnew;
  float x2 = (mix - bn2m[c]) * bn2g[c] * rsqrtf(bn2v[c] + BN_EPS) + bn2b[c];
  out[i] = (h0[i] + x1v + x2) * (1.f / 3.f);
}

// ================================================================ launch
extern "C" void kernel_launch(void* const* d_in, const int* in_sizes, int n_in,
                              void* d_out, int out_size, void* d_ws,
                              size_t ws_size, hipStream_t stream) {
  (void)in_sizes; (void)n_in; (void)out_size; (void)ws_size;
  const float* x = (const float*)d_in[0];
  const int* eidx = (const int*)d_in[1];
  const float* edge_attr = (const float*)d_in[2];
  // d_in[3] = batch (unused)
  const float* w_in = (const float*)d_in[4];
  const float* b_in = (const float*)d_in[5];
  const float* bn_in_g = (const float*)d_in[6];
  const float* bn_in_b = (const float*)d_in[7];
  const float* bn_in_m = (const float*)d_in[8];
  const float* bn_in_v = (const float*)d_in[9];
  const float* gine_ew = (const float*)d_in[10];
  const float* gine_eb = (const float*)d_in[11];
  const float* gine_w1 = (const float*)d_in[12];
  const float* gine_b1 = (const float*)d_in[13];
  const float* gine_bn_g = (const float*)d_in[14];
  const float* gine_bn_b = (const float*)d_in[15];
  const float* gine_bn_m = (const float*)d_in[16];
  const float* gine_bn_v = (const float*)d_in[17];
  const float* gine_w2 = (const float*)d_in[18];
  const float* gine_b2 = (const float*)d_in[19];
  const float* gat_wl = (const float*)d_in[20];
  const float* gat_bl = (const float*)d_in[21];
  const float* gat_wr = (const float*)d_in[22];
  const float* gat_br = (const float*)d_in[23];
  const float* gat_we = (const float*)d_in[24];
  const float* gat_att = (const float*)d_in[25];
  const float* gat_bias = (const float*)d_in[26];
  const float* bn1_g = (const float*)d_in[27];
  const float* bn1_b = (const float*)d_in[28];
  const float* bn1_m = (const float*)d_in[29];
  const float* bn1_v = (const float*)d_in[30];
  const float* bn2_g = (const float*)d_in[31];
  const float* bn2_b = (const float*)d_in[32];
  const float* bn2_m = (const float*)d_in[33];
  const float* bn2_v = (const float*)d_in[34];
  const float* skip0 = (const float*)d_in[35];
  const float* skip1 = (const float*)d_in[36];

  const int* src = eidx;
  const int* dst = eidx + N_EDGES;
  float* out = (float*)d_out;

  // workspace carve-up (floats)
  float* ws = (float*)d_ws;
  float* h0 = ws;      ws += (long)N_NODES * H;
  float* agg = ws;     ws += (long)N_NODES * H;
  float* x1 = ws;      ws += (long)N_NODES * H;
  float* xl = ws;      ws += (long)N_NODES * GAT_C;
  float* xr = ws;      ws += (long)N_NODES * GAT_C;
  float* gatout = ws;  ws += (long)N_NODES * GAT_C;
  float* score = ws;   ws += (long)N_EDGES * HEADS;
  float* mmax = ws;    ws += (long)N_NODES * HEADS;
  float* ssum = ws;    ws += (long)N_NODES * HEADS;

  // per-call inits (deterministic; harness does not re-poison)
  {
    long n;
    n = (long)N_NODES * H;
    fill_kernel<<<(unsigned)((n + 255) / 256), 256, 0, stream>>>(agg, n, 0.f);
    n = (long)N_NODES * GAT_C;
    fill_kernel<<<(unsigned)((n + 255) / 256), 256, 0, stream>>>(gatout, n, 0.f);
    n = (long)N_NODES * HEADS;
    fill_kernel<<<(unsigned)((n + 255) / 256), 256, 0, stream>>>(ssum, n, 0.f);
    fill_kernel<<<(unsigned)((n + 255) / 256), 256, 0, stream>>>(mmax, n, -3.0e38f);
  }

  input_proj_kernel<<<N_NODES / 2, 256, 0, stream>>>(
      x, w_in, b_in, bn_in_g, bn_in_b, bn_in_m, bn_in_v, h0);

  gine_edge_kernel<<<N_EDGES / 2, 256, 0, stream>>>(src, dst, edge_attr,
                                                    gine_ew, gine_eb, h0, agg);

  gine_mlp_kernel<<<N_NODES / 16, 256, 0, stream>>>(
      h0, agg, gine_w1, gine_b1, gine_bn_g, gine_bn_b, gine_bn_m, gine_bn_v,
      gine_w2, gine_b2, bn1_g, bn1_b, bn1_m, bn1_v, skip0, x1);

  gat_transform_kernel<<<N_NODES / 16, 256, 0, stream>>>(x1, gat_wl, gat_bl,
                                                         gat_wr, gat_br, xl, xr);

  gat_score_kernel<<<N_EDGES / 2, 256, 0, stream>>>(
      src, dst, edge_attr, gat_we, gat_att, xl, xr, score, mmax);

  {
    long n = (long)N_EDGES * HEADS;
    gat_expsum_kernel<<<(unsigned)((n + 255) / 256), 256, 0, stream>>>(
        dst, mmax, score, ssum);
  }

  gat_aggregate_kernel<<<N_EDGES / 2, 256, 0, stream>>>(src, dst, score, ssum,
                                                        xl, gatout);

  {
    long n = (long)N_NODES * H;
    finalize_kernel<<<(unsigned)((n + 255) / 256), 256, 0, stream>>>(
        h0, x1, gatout, gat_bias, bn2_g, bn2_b, bn2_m, bn2_v, skip1, out);
  }
}